// GNNEncoder_82540681494885
// MI455X (gfx1250) — compile-verified
//
#include <hip/hip_runtime.h>
#include <hip/hip_bf16.h>

typedef __attribute__((ext_vector_type(2))) float v2f;
typedef __attribute__((ext_vector_type(8))) float v8f;

#define DFEAT 128

__device__ __forceinline__ void atomAddF(float* p, float v) {
    unsafeAtomicAdd(p, v);   // hardware global_atomic_add_f32
}

// ---------------- utility kernels ----------------

__global__ void zero_kernel(float* __restrict__ p, int n) {
    int i = blockIdx.x * blockDim.x + threadIdx.x;
    int stride = gridDim.x * blockDim.x;
    for (; i < n; i += stride) p[i] = 0.0f;
}

__global__ void init_deg_kernel(float* __restrict__ deg, int n) {
    int i = blockIdx.x * blockDim.x + threadIdx.x;
    if (i < n) deg[i] = 1.0f;   // self-loop contributes 1 to in-degree
}

__global__ void add_deg_kernel(const int* __restrict__ dst, float* __restrict__ deg, int nE) {
    int i = blockIdx.x * blockDim.x + threadIdx.x;
    if (i < nE) atomAddF(&deg[dst[i]], 1.0f);
}

__global__ void rsqrt_kernel(float* __restrict__ deg, int n) {
    int i = blockIdx.x * blockDim.x + threadIdx.x;
    if (i < n) deg[i] = __frsqrt_rn(deg[i]);   // deg >= 1 always (self-loop)
}

// ---------------- fp32 WMMA GEMM:  Y[n,128] = X[n,128] @ W[128,128] ----------------
// Block: 256 threads = 8 waves. Block covers 64 rows; wave w owns column tile w*16.
// Per 16x16 output tile: 32 chained V_WMMA_F32_16X16X4_F32 over K=128.
// OOB rows are clamped to row 0 for loads (valid memory, result discarded);
// stores are guarded. This keeps EXEC all-ones through the WMMA chain with no
// saveexec traffic in the inner loop.
__global__ void __launch_bounds__(256)
gemm_wmma_f32(const float* __restrict__ X, const float* __restrict__ W,
              float* __restrict__ Y, int nrows) {
    __shared__ float4 sW4[DFEAT * DFEAT / 4];        // 64 KB
    float* sW = (float*)sW4;

    int tid = threadIdx.x;
    const float4* w4 = (const float4*)W;
    for (int i = tid; i < DFEAT * DFEAT / 4; i += 256) sW4[i] = w4[i];
    __syncthreads();

    int wave = tid >> 5;
    int lane = tid & 31;
    int half = lane >> 4;     // 0: lanes 0-15, 1: lanes 16-31
    int l16  = lane & 15;
    int colBase = wave * 16;

    for (int strip = 0; strip < 4; ++strip) {
        int rowBase = blockIdx.x * 64 + strip * 16;
        int myRow = rowBase + l16;
        int ldRow = myRow < nrows ? myRow : 0;       // clamp, always load
        const float* arow = X + (size_t)ldRow * DFEAT + half * 2;

        v8f acc = {};
        #pragma unroll 8
        for (int k = 0; k < DFEAT; k += 4) {
            // A 16x4 f32: VGPR0 = K=(half*2), VGPR1 = K=(half*2+1), row = l16
            v2f a;
            a.x = arow[k];
            a.y = arow[k + 1];
            // B 4x16 f32: VGPR0 = row K=(half*2), VGPR1 = row K=(half*2+1), col = l16
            v2f b;
            b.x = sW[(k + half * 2) * DFEAT + colBase + l16];
            b.y = sW[(k + half * 2 + 1) * DFEAT + colBase + l16];
            acc = __builtin_amdgcn_wmma_f32_16x16x4_f32(
                false, a, false, b, (short)0, acc, false, false);
        }
        // C/D layout: VGPR r -> M = r + half*8, N = l16
        float* yout = Y + (size_t)(rowBase + half * 8) * DFEAT + colBase + l16;
        if (rowBase + 16 <= nrows) {                 // wave-uniform fast path
            #pragma unroll
            for (int r = 0; r < 8; ++r) yout[(size_t)r * DFEAT] = acc[r];
        } else {
            #pragma unroll
            for (int r = 0; r < 8; ++r) {
                if (rowBase + half * 8 + r < nrows) yout[(size_t)r * DFEAT] = acc[r];
            }
        }
    }
}

// ---------------- edge gather-scale-scatter: out[dst] += dinv[s]*dinv[t]*H[src] ----
// One wave32 per edge; float4 per lane (128 floats = 512B coalesced row).
__global__ void __launch_bounds__(256)
scatter_edges(const float* __restrict__ H, const int* __restrict__ src,
              const int* __restrict__ dst, const float* __restrict__ dinv,
              float* __restrict__ out, int nE) {
    int wave = (blockIdx.x * blockDim.x + threadIdx.x) >> 5;
    int lane = threadIdx.x & 31;
    if (wave >= nE) return;
    int s = src[wave];
    int t = dst[wave];
    float c = dinv[s] * dinv[t];
    const float4 hv = *(const float4*)(H + (size_t)s * DFEAT + lane * 4);
    float* o = out + (size_t)t * DFEAT + lane * 4;
    atomAddF(o + 0, c * hv.x);
    atomAddF(o + 1, c * hv.y);
    atomAddF(o + 2, c * hv.z);
    atomAddF(o + 3, c * hv.w);
}

// ---------------- out = relu(agg + dinv[i]^2 * h + bias) ----------------
__global__ void relu_self_kernel(const float* h, const float* agg,
                                 const float* __restrict__ dinv,
                                 const float* __restrict__ bias,
                                 float* out, int total) {
    int i = blockIdx.x * blockDim.x + threadIdx.x;
    int stride = gridDim.x * blockDim.x;
    for (; i < total; i += stride) {
        int node = i >> 7;
        int d = i & (DFEAT - 1);
        float dv = dinv[node];
        float v = agg[i] + dv * dv * h[i] + bias[d];
        out[i] = fmaxf(v, 0.0f);
    }
}

// ---------------- mean pool: sums[batch[i]] += h[i]; cnts[batch[i]] += 1 ----------
__global__ void __launch_bounds__(256)
pool_kernel(const float* __restrict__ H, const int* __restrict__ batch,
            float* __restrict__ sums, float* __restrict__ cnts, int n) {
    int wave = (blockIdx.x * blockDim.x + threadIdx.x) >> 5;
    int lane = threadIdx.x & 31;
    if (wave >= n) return;
    int g = batch[wave];
    const float4 hv = *(const float4*)(H + (size_t)wave * DFEAT + lane * 4);
    float* o = sums + (size_t)g * DFEAT + lane * 4;
    atomAddF(o + 0, hv.x);
    atomAddF(o + 1, hv.y);
    atomAddF(o + 2, hv.z);
    atomAddF(o + 3, hv.w);
    if (lane == 0) atomAddF(&cnts[g], 1.0f);
}

__global__ void finalize_kernel(float* __restrict__ out, const float* __restrict__ cnts,
                                int total) {
    int i = blockIdx.x * blockDim.x + threadIdx.x;
    if (i < total) out[i] = out[i] / fmaxf(cnts[i >> 7], 1.0f);
}

// ---------------- launcher ----------------

static inline int cdiv(int a, int b) { return (a + b - 1) / b; }

extern "C" void kernel_launch(void* const* d_in, const int* in_sizes, int n_in,
                              void* d_out, int out_size, void* d_ws, size_t ws_size,
                              hipStream_t stream) {
    const float* x  = (const float*)d_in[0];
    const float* W1 = (const float*)d_in[1];
    const float* b1 = (const float*)d_in[2];
    const float* W2 = (const float*)d_in[3];
    const float* b2 = (const float*)d_in[4];
    const int* edge  = (const int*)d_in[5];
    const int* batch = (const int*)d_in[6];

    int n_nodes = in_sizes[0] / DFEAT;
    int n_edges = in_sizes[5] / 2;
    const int* src = edge;
    const int* dst = edge + n_edges;

    float* out = (float*)d_out;

    // workspace layout
    float* dinv = (float*)d_ws;
    size_t nAlign = ((size_t)n_nodes + 255) & ~(size_t)255;
    float* bufA = dinv + nAlign;
    float* bufB = bufA + (size_t)n_nodes * DFEAT;
    float* cnts = bufB + (size_t)n_nodes * DFEAT;

    int total = n_nodes * DFEAT;
    dim3 blk(256);

    // --- symmetric normalization coefficients ---
    init_deg_kernel<<<cdiv(n_nodes, 256), blk, 0, stream>>>(dinv, n_nodes);
    add_deg_kernel<<<cdiv(n_edges, 256), blk, 0, stream>>>(dst, dinv, n_edges);
    rsqrt_kernel<<<cdiv(n_nodes, 256), blk, 0, stream>>>(dinv, n_nodes);

    int gemmBlocks = cdiv(n_nodes, 64);
    int edgeBlocks = cdiv(n_edges, 8);   // 8 waves per block, 1 wave per edge
    int nodeBlocks = cdiv(n_nodes, 8);

    // --- layer 1 ---
    gemm_wmma_f32<<<gemmBlocks, blk, 0, stream>>>(x, W1, bufA, n_nodes);
    zero_kernel<<<cdiv(total, 1024), blk, 0, stream>>>(bufB, total);
    scatter_edges<<<edgeBlocks, blk, 0, stream>>>(bufA, src, dst, dinv, bufB, n_edges);
    relu_self_kernel<<<cdiv(total, 1024), blk, 0, stream>>>(bufA, bufB, dinv, b1, bufA, total);

    // --- layer 2 ---
    gemm_wmma_f32<<<gemmBlocks, blk, 0, stream>>>(bufA, W2, bufB, n_nodes);
    zero_kernel<<<cdiv(total, 1024), blk, 0, stream>>>(bufA, total);
    scatter_edges<<<edgeBlocks, blk, 0, stream>>>(bufB, src, dst, dinv, bufA, n_edges);
    relu_self_kernel<<<cdiv(total, 1024), blk, 0, stream>>>(bufB, bufA, dinv, b2, bufB, total);

    // --- global mean pool ---
    zero_kernel<<<cdiv(out_size, 256), blk, 0, stream>>>(out, out_size);
    zero_kernel<<<1, 64, 0, stream>>>(cnts, 64);
    pool_kernel<<<nodeBlocks, blk, 0, stream>>>(bufB, batch, out, cnts, n_nodes);
    finalize_kernel<<<cdiv(out_size, 256), blk, 0, stream>>>(out, cnts, out_size);
}